// CA3RecurrentMatrix_45243185496623
// MI455X (gfx1250) — compile-verified
//
#include <hip/hip_runtime.h>
#include <math.h>

typedef __attribute__((ext_vector_type(2))) float v2f;
typedef __attribute__((ext_vector_type(8))) float v8f;
typedef int b128_t __attribute__((vector_size(16)));   // matches async-builtin param type

#define AS1 __attribute__((address_space(1)))
#define AS3 __attribute__((address_space(3)))

#if defined(__AMDGCN__) && __has_builtin(__builtin_amdgcn_global_load_async_to_lds_b128)
#define USE_ASYNC_LDS 1
#else
#define USE_ASYNC_LDS 0
#endif

#if USE_ASYNC_LDS
#if __has_builtin(__builtin_amdgcn_s_wait_asynccnt)
#define WAIT_ASYNC() __builtin_amdgcn_s_wait_asynccnt(0)
#else
#define WAIT_ASYNC() asm volatile("s_wait_asynccnt 0x0" ::: "memory")
#endif
#else
#define WAIT_ASYNC() ((void)0)
#endif

namespace {

constexpr int KSLOTS = 4096;   // ca3_memory_slots
constexpr int CDIM   = 2048;   // ca3_code_dim
constexpr int BQ     = 8192;   // query batch
constexpr int MAX_ITERS = 8;
constexpr float TOL = 1e-4f;
constexpr float ALPHA_CLAMP = 5e-4f;

constexpr int BM = 128, BN = 128, BK = 32;
constexpr int LDPA = 36;   // A tile [m][k] padded row stride: conflict-free b64 frag loads
constexpr int LDPB = 36;   // B tile stored transposed [n][k], same stride/padding

enum { EP_STORE = 0, EP_UPDATE = 1, EP_RESID = 2 };

// A tile 128x32 row-major -> LDS, via async global->LDS DMA (ASYNCcnt)
__device__ __forceinline__ void issue_a_tile(
    float* __restrict__ Ad, const float* __restrict__ Am,
    int Kd, int m0, int kb, int tid)
{
#if USE_ASYNC_LDS
#pragma unroll
    for (int i = 0; i < 4; ++i) {
        const int f4 = tid + (i << 8);
        const int r  = f4 >> 3;            // 0..127
        const int c  = (f4 & 7) << 2;      // 0..28
        __builtin_amdgcn_global_load_async_to_lds_b128(
            (AS1 b128_t*)(Am + (size_t)(m0 + r) * Kd + kb + c),
            (AS3 b128_t*)(Ad + r * LDPA + c), 0, 0);
    }
#else
#pragma unroll
    for (int i = 0; i < 4; ++i) {
        const int f4 = tid + (i << 8);
        const int r  = f4 >> 3;
        const int c  = (f4 & 7) << 2;
        *(float4*)(Ad + r * LDPA + c) =
            *(const float4*)(Am + (size_t)(m0 + r) * Kd + kb + c);
    }
#endif
}

// B tile 32x128: global -> registers (issued early, waited late)
__device__ __forceinline__ void fetch_b_tile(
    float4* __restrict__ bv, const float* __restrict__ Bm,
    int N, int n0, int kb, int tid)
{
#pragma unroll
    for (int i = 0; i < 4; ++i) {
        const int f4 = tid + (i << 8);
        bv[i] = *(const float4*)(Bm + (size_t)(kb + (f4 >> 5)) * N + n0 + ((f4 & 31) << 2));
    }
}

// registers -> LDS, transposed to [n][k] so frag loads are contiguous b64
__device__ __forceinline__ void store_b_tile(
    float* __restrict__ Bt, const float4* __restrict__ bv, int tid)
{
#pragma unroll
    for (int i = 0; i < 4; ++i) {
        const int f4 = tid + (i << 8);
        const int r  = f4 >> 5;            // 0..31  (k)
        const int c  = (f4 & 31) << 2;     // 0..124 (n)
        Bt[(c + 0) * LDPB + r] = bv[i].x;
        Bt[(c + 1) * LDPB + r] = bv[i].y;
        Bt[(c + 2) * LDPB + r] = bv[i].z;
        Bt[(c + 3) * LDPB + r] = bv[i].w;
    }
}

// D[M,N] = A[M,Kd] @ B[Kd,N] (row-major, dims multiples of 128) with epilogue:
//   EP_STORE : D = acc
//   EP_UPDATE: D = done ? Px : 2*Px - acc          (Ben-Israel masked update)
//   EP_RESID : partials[bid] = sum((acc - Px)^2)   (Frobenius residual, no D store)
template <int MODE>
__global__ __launch_bounds__(256) void gemm_wmma(
    const float* __restrict__ Am, const float* __restrict__ Bm,
    float* __restrict__ D,
    const float* __restrict__ Px,
    const int* __restrict__ doneFlag,
    float* __restrict__ partials,
    int M, int N, int Kd)
{
    __shared__ float AsSh[2][BM * LDPA];   // A tiles [m][k], double-buffered
    __shared__ float BsSh[2][BN * LDPB];   // B tiles transposed [n][k], double-buffered

    const int tid  = threadIdx.x;
    const int lane = tid & 31;
    const int wid  = tid >> 5;
    const int wm   = wid >> 1;             // 0..3 : 32-row band of wave
    const int wn   = wid & 1;              // 0..1 : 64-col band of wave
    const int ml   = lane & 15;            // M/N index within 16x16 tile
    const int kl   = (lane >> 4) << 1;     // K sub-offset per ISA f32 operand layout (0 or 2)
    const int m0   = blockIdx.y * BM;
    const int n0   = blockIdx.x * BN;

    v8f acc[2][4];
#pragma unroll
    for (int i = 0; i < 2; ++i)
#pragma unroll
        for (int j = 0; j < 4; ++j)
#pragma unroll
            for (int r = 0; r < 8; ++r) acc[i][j][r] = 0.0f;

    // Prologue: fill buffer 0
    float4 bv[4];
    issue_a_tile(AsSh[0], Am, Kd, m0, 0, tid);
    fetch_b_tile(bv, Bm, N, n0, 0, tid);
    store_b_tile(BsSh[0], bv, tid);
    WAIT_ASYNC();
    __syncthreads();

    int parity = 0;
    for (int kb = 0; kb < Kd; kb += BK) {
        const float* Ac = AsSh[parity];
        const float* Bt = BsSh[parity];
        const bool hasNext = (kb + BK) < Kd;
        // Stream next tile: async A -> other LDS buffer; B -> registers (wait deferred)
        if (hasNext) {
            issue_a_tile(AsSh[parity ^ 1], Am, Kd, m0, kb + BK, tid);
            fetch_b_tile(bv, Bm, N, n0, kb + BK, tid);
        }

        // 8 k-steps of V_WMMA_F32_16X16X4_F32 per BK chunk
#pragma unroll
        for (int k4 = 0; k4 < 8; ++k4) {
            const int kk = (k4 << 2) + kl;
            const v2f a0 = *(const v2f*)(Ac + (wm * 32 +      ml) * LDPA + kk);
            const v2f a1 = *(const v2f*)(Ac + (wm * 32 + 16 + ml) * LDPA + kk);
            v2f b[4];
#pragma unroll
            for (int j = 0; j < 4; ++j)
                b[j] = *(const v2f*)(Bt + (wn * 64 + j * 16 + ml) * LDPB + kk);
#pragma unroll
            for (int j = 0; j < 4; ++j) {
                acc[0][j] = __builtin_amdgcn_wmma_f32_16x16x4_f32(
                    false, a0, false, b[j], (short)0, acc[0][j], false, false);
                acc[1][j] = __builtin_amdgcn_wmma_f32_16x16x4_f32(
                    false, a1, false, b[j], (short)0, acc[1][j], false, false);
            }
        }
        // Scatter staged B registers into the other buffer (loadcnt wait lands here)
        if (hasNext) store_b_tile(BsSh[parity ^ 1], bv, tid);
        WAIT_ASYNC();        // next A tile fully landed in LDS
        __syncthreads();     // all waves done reading current buffers / stores visible
        parity ^= 1;
    }

    // Epilogue. D layout per ISA: v8f element r -> row = r + 8*(lane>>4), col = lane&15
    const int roff = (lane >> 4) << 3;
    float lsum = 0.0f;
    bool dn = false;
    if (MODE == EP_UPDATE) dn = (*doneFlag != 0);
#pragma unroll
    for (int im = 0; im < 2; ++im)
#pragma unroll
        for (int jn = 0; jn < 4; ++jn)
#pragma unroll
            for (int r = 0; r < 8; ++r) {
                const int row = m0 + wm * 32 + im * 16 + roff + r;
                const int col = n0 + wn * 64 + jn * 16 + ml;
                const size_t idx = (size_t)row * N + col;
                const float v = acc[im][jn][r];
                if (MODE == EP_STORE) {
                    D[idx] = v;
                } else if (MODE == EP_UPDATE) {
                    const float p = Px[idx];
                    D[idx] = dn ? p : (2.0f * p - v);
                } else {
                    const float d = v - Px[idx];
                    lsum += d * d;
                }
            }
    if (MODE == EP_RESID) {
        float* red = &AsSh[0][0];
        __syncthreads();
        red[tid] = lsum;
        __syncthreads();
#pragma unroll
        for (int s = 128; s > 0; s >>= 1) {
            if (tid < s) red[tid] += red[tid + s];
            __syncthreads();
        }
        if (tid == 0) partials[blockIdx.y * gridDim.x + blockIdx.x] = red[0];
    }
}

// Deterministic ||A||_F^2 partials (no float atomics)
__global__ __launch_bounds__(256) void frob_partials(
    const float* __restrict__ A, float* __restrict__ part, int n4)
{
    __shared__ float sm[256];
    float s = 0.0f;
    const float4* A4 = (const float4*)A;
    for (int i = blockIdx.x * 256 + threadIdx.x; i < n4; i += gridDim.x * 256) {
        const float4 v = A4[i];
        s += v.x * v.x + v.y * v.y + v.z * v.z + v.w * v.w;
    }
    sm[threadIdx.x] = s;
    __syncthreads();
#pragma unroll
    for (int t = 128; t > 0; t >>= 1) {
        if (threadIdx.x < t) sm[threadIdx.x] += sm[threadIdx.x + t];
        __syncthreads();
    }
    if (threadIdx.x == 0) part[blockIdx.x] = sm[0];
}

__global__ void alpha_init(const float* __restrict__ part, int np,
                           const float* __restrict__ log_scale,
                           float* __restrict__ alpha, int* __restrict__ done)
{
    if (blockIdx.x == 0 && threadIdx.x == 0) {
        float s = 0.0f;
        for (int i = 0; i < np; ++i) s += part[i];
        *alpha = fminf(expf(*log_scale), ALPHA_CLAMP) / (s + 1e-8f);
        *done = 0;
    }
}

// P[c][k] = alpha * A[k][c]   (pinv init = alpha * A^T)
__global__ __launch_bounds__(256) void transpose_scale(
    const float* __restrict__ A, float* __restrict__ P,
    const float* __restrict__ alpha, int Krows, int Ccols)
{
    __shared__ float t[32][33];
    const float a = *alpha;
    const int c0 = blockIdx.x * 32, k0 = blockIdx.y * 32;
    const int x = threadIdx.x & 31, y = threadIdx.x >> 5;   // 32 x 8
#pragma unroll
    for (int j = 0; j < 4; ++j)
        t[y + j * 8][x] = A[(size_t)(k0 + y + j * 8) * Ccols + c0 + x];
    __syncthreads();
#pragma unroll
    for (int j = 0; j < 4; ++j)
        P[(size_t)(c0 + y + j * 8) * Krows + k0 + x] = a * t[x][y + j * 8];
}

__global__ void res_finish(const float* __restrict__ part, int np, int* __restrict__ done)
{
    if (blockIdx.x == 0 && threadIdx.x == 0) {
        float s = 0.0f;
        for (int i = 0; i < np; ++i) s += part[i];
        if (sqrtf(s) < TOL) *done = 1;   // done |= (residual < TOL)
    }
}

} // namespace

extern "C" void kernel_launch(void* const* d_in, const int* in_sizes, int n_in,
                              void* d_out, int out_size, void* d_ws, size_t ws_size,
                              hipStream_t stream)
{
    (void)in_sizes; (void)n_in; (void)out_size; (void)ws_size;

    const float* query = (const float*)d_in[0];   // (B, C)
    const float* Amat  = (const float*)d_in[1];   // (K, C)
    const float* logsc = (const float*)d_in[2];   // scalar
    float* out = (float*)d_out;                   // (B, C)

    // Workspace carve-out (256B aligned): 2x pinv (C*K) + 2x T (C*C) + partials + scalars
    char* w = (char*)d_ws;
    auto take = [&](size_t bytes) -> void* {
        void* p = (void*)w;
        w += (bytes + 255) & ~(size_t)255;
        return p;
    };
    float* pinvA = (float*)take((size_t)CDIM * KSLOTS * sizeof(float));
    float* pinvB = (float*)take((size_t)CDIM * KSLOTS * sizeof(float));
    float* Ta    = (float*)take((size_t)CDIM * CDIM * sizeof(float));
    float* Tb    = (float*)take((size_t)CDIM * CDIM * sizeof(float));
    float* part  = (float*)take(1024 * sizeof(float));
    float* alpha = (float*)take(256);
    int*   done  = (int*)take(256);

    // alpha = min(exp(log_scale), 5e-4) / (||A||_F^2 + 1e-8) ; done = 0
    frob_partials<<<512, 256, 0, stream>>>(Amat, part, (KSLOTS * CDIM) / 4);
    alpha_init<<<1, 1, 0, stream>>>(part, 512, logsc, alpha, done);

    // pinv_0 = alpha * A^T
    transpose_scale<<<dim3(CDIM / 32, KSLOTS / 32), 256, 0, stream>>>(Amat, pinvA, alpha, KSLOTS, CDIM);

    // T_0 = pinv_0 @ A  (C x C) — reused both for the update and the residual
    gemm_wmma<EP_STORE><<<dim3(CDIM / BN, CDIM / BM), 256, 0, stream>>>(
        pinvA, Amat, Ta, nullptr, nullptr, nullptr, CDIM, CDIM, KSLOTS);

    float* pc = pinvA; float* pn = pinvB;
    float* Tc = Ta;    float* Tn = Tb;
    for (int it = 0; it < MAX_ITERS; ++it) {
        // pinv_next = done ? pinv : 2*pinv - (pinv@A)@pinv      [ (C,C)@(C,K) ]
        gemm_wmma<EP_UPDATE><<<dim3(KSLOTS / BN, CDIM / BM), 256, 0, stream>>>(
            Tc, pc, pn, pc, done, nullptr, CDIM, KSLOTS, CDIM);
        // T_next = pinv_next @ A                                [ (C,K)@(K,C) ]
        gemm_wmma<EP_STORE><<<dim3(CDIM / BN, CDIM / BM), 256, 0, stream>>>(
            pn, Amat, Tn, nullptr, nullptr, nullptr, CDIM, CDIM, KSLOTS);
        // residual = ||A @ T_next - A||_F                       [ (K,C)@(C,C) ]
        gemm_wmma<EP_RESID><<<dim3(CDIM / BN, KSLOTS / BM), 256, 0, stream>>>(
            Amat, Tn, nullptr, Amat, nullptr, part, KSLOTS, CDIM, CDIM);
        res_finish<<<1, 1, 0, stream>>>(part, (CDIM / BN) * (KSLOTS / BM), done);
        { float* t = pc; pc = pn; pn = t; }
        { float* t = Tc; Tc = Tn; Tn = t; }
    }

    // retrieved = query @ (pinv_final @ A) = query @ T_final    [ (B,C)@(C,C) ]
    gemm_wmma<EP_STORE><<<dim3(CDIM / BN, BQ / BM), 256, 0, stream>>>(
        query, Tc, out, nullptr, nullptr, nullptr, BQ, CDIM, CDIM);
}